// MyNormScanSali_68436008894677
// MI455X (gfx1250) — compile-verified
//
#include <hip/hip_runtime.h>
#include <hip/hip_bf16.h>
#include <math.h>

// Problem constants (from the reference): B=128, H=480, W=640.
constexpr int BIMG  = 128;
constexpr int HW    = 480 * 640;      // 307200 floats per image
constexpr int N4    = HW / 4;         // 76800 float4 per image
constexpr int SPLIT = 12;             // segments per image
constexpr int CHUNK4 = N4 / SPLIT;    // 6400 float4 per block (exact)
constexpr int TPB   = 256;            // 8 wave32s per block

typedef float  v4f __attribute__((ext_vector_type(4)));
typedef float  v2f __attribute__((ext_vector_type(2)));
typedef float  v8f __attribute__((ext_vector_type(8)));

// ---------------------------------------------------------------------------
// Stage 1: streaming partial sums. One block handles one (image, segment)
// chunk of 6400 float4 (exactly 25 iterations of 256 threads -> uniform,
// fully unrollable, remainder-free). Non-temporal b128 loads: single-use
// 315 MB stream, don't cache it.
// Writes 4 exact partials per block -> deterministic (no float atomics).
// ws layout: P[type][image][segment], type in {sum, sumsq, summask, sumxm}.
// ---------------------------------------------------------------------------
__global__ void __launch_bounds__(TPB)
nss_partial_kernel(const float* __restrict__ x,
                   const float* __restrict__ t,
                   float* __restrict__ ws) {
    const int bid = blockIdx.x;
    const int img = bid / SPLIT;
    const int seg = bid % SPLIT;

    const v4f* __restrict__ xp =
        (const v4f*)(x + (size_t)img * HW) + (size_t)seg * CHUNK4;
    const v4f* __restrict__ tp =
        (const v4f*)(t + (size_t)img * HW) + (size_t)seg * CHUNK4;

    float s = 0.f, ss = 0.f, sm = 0.f, sxm = 0.f;

#pragma unroll 5
    for (int i = threadIdx.x; i < CHUNK4; i += TPB) {
        v4f xv = __builtin_nontemporal_load(xp + i);   // global_load_b128, TH=NT
        v4f tv = __builtin_nontemporal_load(tp + i);
#pragma unroll
        for (int c = 0; c < 4; ++c) {
            float xe = xv[c];
            float m  = (tv[c] != 0.0f) ? 1.0f : 0.0f;  // v_cmp + v_cndmask
            s   += xe;
            ss   = fmaf(xe, xe, ss);
            sm  += m;
            sxm  = fmaf(m, xe, sxm);
        }
    }

    // Wave32 butterfly reduction (fixed order -> deterministic).
#pragma unroll
    for (int msk = 16; msk >= 1; msk >>= 1) {
        s   += __shfl_xor(s,   msk, 32);
        ss  += __shfl_xor(ss,  msk, 32);
        sm  += __shfl_xor(sm,  msk, 32);
        sxm += __shfl_xor(sxm, msk, 32);
    }

    __shared__ float red[4][TPB / 32];
    const int wave = threadIdx.x >> 5;
    const int lane = threadIdx.x & 31;
    if (lane == 0) {
        red[0][wave] = s; red[1][wave] = ss; red[2][wave] = sm; red[3][wave] = sxm;
    }
    __syncthreads();
    if (threadIdx.x == 0) {
        float a0 = 0.f, a1 = 0.f, a2 = 0.f, a3 = 0.f;
#pragma unroll
        for (int w = 0; w < TPB / 32; ++w) {
            a0 += red[0][w]; a1 += red[1][w]; a2 += red[2][w]; a3 += red[3][w];
        }
        ws[(0 * BIMG + img) * SPLIT + seg] = a0;
        ws[(1 * BIMG + img) * SPLIT + seg] = a1;
        ws[(2 * BIMG + img) * SPLIT + seg] = a2;
        ws[(3 * BIMG + img) * SPLIT + seg] = a3;
    }
}

// ---------------------------------------------------------------------------
// Stage 2: P[type][128][12] x ones(12) via V_WMMA_F32_16X16X4_F32, then the
// scalar NSS finalize. Single wave32, all 32 lanes active (EXEC all-ones, as
// WMMA requires). Full f32 precision (sums reach ~3e5; f16/bf16 would not do).
//
// A-layout (ISA 7.12.2, 32-bit A 16x4): lanes 0-15 hold K=0 (VGPR0) / K=1
// (VGPR1); lanes 16-31 hold K=2 / K=3. M = lane%16. B = all-ones, which is
// layout-independent. After 3 K-steps (12 segments), D[m][n] = sum_s P[m][s]
// for every n; D-layout: VGPR r = rows {r, r+8} on the two lane halves.
// ---------------------------------------------------------------------------
__global__ void __launch_bounds__(32)
nss_finalize_kernel(const float* __restrict__ ws, float* __restrict__ out) {
    __shared__ float S[4][BIMG];

    const int L    = threadIdx.x;        // 0..31, all active
    const int half = (L >= 16) ? 1 : 0;  // which K-half this lane supplies
    const int mrow = L & 15;             // matrix row M

    v2f bones;
    bones[0] = 1.0f; bones[1] = 1.0f;

    for (int tile = 0; tile < BIMG / 16; ++tile) {      // 8 image tiles
        for (int tau = 0; tau < 4; ++tau) {             // 4 partial types
            const float* base = ws + (tau * BIMG + tile * 16 + mrow) * SPLIT;
            v8f acc = {};
#pragma unroll
            for (int ks = 0; ks < SPLIT / 4; ++ks) {    // 3 K-steps of 4
                const int k0 = ks * 4 + half * 2;
                v2f a;
                a[0] = base[k0];
                a[1] = base[k0 + 1];
                acc = __builtin_amdgcn_wmma_f32_16x16x4_f32(
                    /*neg_a=*/false, a, /*neg_b=*/false, bones,
                    /*c_mod=*/(short)0, acc,
                    /*reuse_a=*/false, /*reuse_b=*/false);
            }
            // Every column of row m is identical; lanes with N==0 extract.
            if (mrow == 0) {
#pragma unroll
                for (int r = 0; r < 8; ++r)
                    S[tau][tile * 16 + r + half * 8] = acc[r];
            }
        }
    }
    __syncthreads();

    const float Nf = (float)HW;
    float accn = 0.f;
#pragma unroll
    for (int q = 0; q < BIMG / 32; ++q) {   // 4 images per lane
        const int img = L + q * 32;
        const float s   = S[0][img];
        const float ss  = S[1][img];
        const float sm  = S[2][img];
        const float sxm = S[3][img];
        const float mean = s / Nf;
        const float var  = (ss - s * s / Nf) / (Nf - 1.0f);   // ddof=1
        const float stdv = sqrtf(var);
        const float nss  = (sxm - mean * sm) / (stdv * Nf);
        accn += nss;
    }
#pragma unroll
    for (int msk = 16; msk >= 1; msk >>= 1)
        accn += __shfl_xor(accn, msk, 32);
    if (L == 0)
        out[0] = accn / (float)BIMG;
}

// ---------------------------------------------------------------------------
extern "C" void kernel_launch(void* const* d_in, const int* in_sizes, int n_in,
                              void* d_out, int out_size, void* d_ws, size_t ws_size,
                              hipStream_t stream) {
    const float* x = (const float*)d_in[0];   // input  [B,H,W] f32
    const float* t = (const float*)d_in[1];   // target [B,H,W] f32
    float* out = (float*)d_out;               // scalar f32
    float* ws  = (float*)d_ws;                // needs 4*128*12*4 = 24 KB

    (void)in_sizes; (void)n_in; (void)out_size; (void)ws_size;

    nss_partial_kernel<<<dim3(BIMG * SPLIT), dim3(TPB), 0, stream>>>(x, t, ws);
    nss_finalize_kernel<<<dim3(1), dim3(32), 0, stream>>>(ws, out);
}